// OrthogonalIntegrityAxiom_17583596110438
// MI455X (gfx1250) — compile-verified
//
#include <hip/hip_runtime.h>
#include <hip/hip_bf16.h>

#define EPS_F 1e-8f

typedef float v2f __attribute__((ext_vector_type(2)));
typedef float v8f __attribute__((ext_vector_type(8)));

// ---------------- workspace zeroing ----------------
__global__ void oia_zero_ws(float* __restrict__ ws, int nwords) {
    int i = blockIdx.x * blockDim.x + threadIdx.x;
    if (i < nwords) ws[i] = 0.0f;
}

// ---------------- edge scatter: per-node {C4, S4, unit-count, total-count} ----------------
__global__ void oia_edge_accum(const float* __restrict__ pos,
                               const int* __restrict__ src,
                               const int* __restrict__ dst,
                               float* __restrict__ C4, float* __restrict__ S4,
                               float* __restrict__ DU, float* __restrict__ DT,
                               int E) {
    int e = blockIdx.x * blockDim.x + threadIdx.x;
    if (e >= E) return;
    int u = src[e];
    int v = dst[e];
    float2 pu = ((const float2*)pos)[u];
    float2 pv = ((const float2*)pos)[v];
    float dx = pv.x - pu.x;
    float dy = pv.y - pu.y;
    float r = sqrtf(fmaf(dx, dx, dy * dy));
    float w = (r >= EPS_F) ? 1.0f : 0.0f;       // unit-vector slot (self-loops -> 0)
    float inv = w / fmaxf(r, EPS_F);
    float c = dx * inv, s = dy * inv;            // cos(theta), sin(theta) (or 0,0)
    float c2 = (c - s) * (c + s);                // cos(2theta)
    float s2 = 2.0f * c * s;                     // sin(2theta)
    float c4 = (c2 - s2) * (c2 + s2);            // cos(4theta)
    float s4 = 2.0f * c2 * s2;                   // sin(4theta)
    // both endpoint slots see the same edge direction d
    unsafeAtomicAdd(&C4[u], c4);
    unsafeAtomicAdd(&S4[u], s4);
    unsafeAtomicAdd(&DU[u], w);
    unsafeAtomicAdd(&DT[u], 1.0f);
    unsafeAtomicAdd(&C4[v], c4);
    unsafeAtomicAdd(&S4[v], s4);
    unsafeAtomicAdd(&DU[v], w);
    unsafeAtomicAdd(&DT[v], 1.0f);
}

// ---------------- node pass: per-node closed form + WMMA wave reduction ----------------
__global__ void oia_node_reduce(const float* __restrict__ C4, const float* __restrict__ S4,
                                const float* __restrict__ DU, const float* __restrict__ DT,
                                float* __restrict__ acc, int N) {
    int i = blockIdx.x * blockDim.x + threadIdx.x;
    float num = 0.0f;   // n_u^2 - C4^2 - S4^2   (x16 pair-sum numerator)
    float pcnt = 0.0f;  // n_t*(n_t-1)/2         (pair count)
    if (i < N) {
        float c = C4[i];
        float s = S4[i];
        float nu = DU[i];
        float nt = DT[i];
        num  = fmaf(nu, nu, -fmaf(c, c, s * s));
        pcnt = 0.5f * nt * (nt - 1.0f);
    }

    // Exact f32 wave32 reduction via V_WMMA_F32_16X16X4_F32.
    // B (4x16): vgpr0 = {K0 | K2}, vgpr1 = {K1 | K3} -> b.x=num, b.y=pcnt packs
    //   B[0][n]=num_n, B[1][n]=p_n, B[2][n]=num_{n+16}, B[3][n]=p_{n+16}.
    // A (16x4) selector rows: row0 = (1,0,1,0) sums num; row1 = (0,1,0,1) sums pcnt.
    // D[0][n] = num_n + num_{n+16} (lanes 0-15 of vgpr0); D[1][n] likewise for pcnt.
    int lane = threadIdx.x & 31;
    v2f a;
    a.x = ((lane & 15) == 0) ? 1.0f : 0.0f;   // A[0][0], A[0][2]
    a.y = ((lane & 15) == 1) ? 1.0f : 0.0f;   // A[1][1], A[1][3]
    v2f b;
    b.x = num;
    b.y = pcnt;
    v8f cz = {};
    v8f d = __builtin_amdgcn_wmma_f32_16x16x4_f32(false, a, false, b, (short)0, cz,
                                                  false, false);
    float rn = d[0];   // lanes 0-15: column partial sums of num; lanes 16-31: 0
    float rp = d[1];   // lanes 0-15: column partial sums of pcnt; lanes 16-31: 0
    #pragma unroll
    for (int m = 1; m <= 8; m <<= 1) {
        rn += __shfl_xor(rn, m, 32);
        rp += __shfl_xor(rp, m, 32);
    }
    // lane 0 of each wave now holds the full 32-lane sums
    __shared__ float sn[8];
    __shared__ float sp[8];
    int wave = threadIdx.x >> 5;
    if (lane == 0) { sn[wave] = rn; sp[wave] = rp; }
    __syncthreads();
    if (threadIdx.x == 0) {
        float tn = 0.0f, tp = 0.0f;
        int nw = (blockDim.x + 31) >> 5;
        for (int k = 0; k < nw; ++k) { tn += sn[k]; tp += sp[k]; }
        unsafeAtomicAdd(&acc[0], tn);
        unsafeAtomicAdd(&acc[1], tp);
    }
}

// ---------------- finalize ----------------
__global__ void oia_finalize(const float* __restrict__ acc, float* __restrict__ out) {
    float p = acc[1];
    out[0] = (p > 0.0f) ? acc[0] / (16.0f * p) : 0.0f;
}

extern "C" void kernel_launch(void* const* d_in, const int* in_sizes, int n_in,
                              void* d_out, int out_size, void* d_ws, size_t ws_size,
                              hipStream_t stream) {
    const float* pos = (const float*)d_in[0];   // (1, N, 2) f32
    const int* ei    = (const int*)d_in[2];     // (2, E) i32
    int N = in_sizes[0] / 2;
    int E = in_sizes[2] / 2;

    float* ws  = (float*)d_ws;
    float* acc = ws;            // [0]=numerator sum, [1]=pair count
    float* C4  = ws + 2;
    float* S4  = C4 + N;
    float* DU  = S4 + N;
    float* DT  = DU + N;

    int nwords = 2 + 4 * N;
    oia_zero_ws<<<(nwords + 255) / 256, 256, 0, stream>>>(ws, nwords);
    oia_edge_accum<<<(E + 255) / 256, 256, 0, stream>>>(pos, ei, ei + E,
                                                        C4, S4, DU, DT, E);
    oia_node_reduce<<<(N + 255) / 256, 256, 0, stream>>>(C4, S4, DU, DT, acc, N);
    oia_finalize<<<1, 1, 0, stream>>>(acc, (float*)d_out);
}